// QCNN__gen244_65481071406714
// MI455X (gfx1250) — compile-verified
//
#include <hip/hip_runtime.h>
#include <math.h>

typedef float v2f __attribute__((ext_vector_type(2)));
typedef float v8f __attribute__((ext_vector_type(8)));

#define SEQ 128

// packed weight offsets in LDS
#define OFF_WFM 0      // 16x8
#define OFF_BFM 128
#define OFF_WC1 144    // 16x16
#define OFF_BC1 400
#define OFF_WP1 416    // 12x16
#define OFF_BP1 608
#define OFF_WC2 620    // 8x12
#define OFF_BC2 716
#define OFF_WP2 724    // 4x8
#define OFF_BP2 756
#define OFF_WC3 760    // 4x4
#define OFF_BC3 776
#define OFF_WQ  780    // 4x4
#define OFF_WK  796
#define OFF_WV  812
#define OFF_WFC 828    // 1x4
#define OFF_BFC 832
#define W_TOTAL 833

__device__ __forceinline__ void cpy_lds(float* dst, const float* __restrict__ src,
                                        int n, int tid, int nth) {
    for (int i = tid; i < n; i += nth) dst[i] = src[i];
}

__global__ __launch_bounds__(256)
void qcnn_fused_kernel(const float* __restrict__ x,
                       const float* __restrict__ w_fm, const float* __restrict__ b_fm,
                       const float* __restrict__ w_c1, const float* __restrict__ b_c1,
                       const float* __restrict__ w_p1, const float* __restrict__ b_p1,
                       const float* __restrict__ w_c2, const float* __restrict__ b_c2,
                       const float* __restrict__ w_p2, const float* __restrict__ b_p2,
                       const float* __restrict__ w_c3, const float* __restrict__ b_c3,
                       const float* __restrict__ w_q,  const float* __restrict__ w_k,
                       const float* __restrict__ w_v,
                       const float* __restrict__ w_fc, const float* __restrict__ b_fc,
                       float* __restrict__ out) {
    __shared__ float w[W_TOTAL];
    __shared__ float qs[SEQ][4];
    __shared__ float ks[SEQ][4];
    __shared__ float vs[SEQ][4];
    __shared__ float sc[SEQ][SEQ + 1];   // padded: conflict-free row sweeps

    const int b   = blockIdx.x;
    const int tid = threadIdx.x;
    const int nth = blockDim.x;          // 256

    // ---- stage weights into LDS ----
    cpy_lds(w + OFF_WFM, w_fm, 128, tid, nth);
    cpy_lds(w + OFF_BFM, b_fm, 16,  tid, nth);
    cpy_lds(w + OFF_WC1, w_c1, 256, tid, nth);
    cpy_lds(w + OFF_BC1, b_c1, 16,  tid, nth);
    cpy_lds(w + OFF_WP1, w_p1, 192, tid, nth);
    cpy_lds(w + OFF_BP1, b_p1, 12,  tid, nth);
    cpy_lds(w + OFF_WC2, w_c2, 96,  tid, nth);
    cpy_lds(w + OFF_BC2, b_c2, 8,   tid, nth);
    cpy_lds(w + OFF_WP2, w_p2, 32,  tid, nth);
    cpy_lds(w + OFF_BP2, b_p2, 4,   tid, nth);
    cpy_lds(w + OFF_WC3, w_c3, 16,  tid, nth);
    cpy_lds(w + OFF_BC3, b_c3, 4,   tid, nth);
    cpy_lds(w + OFF_WQ,  w_q,  16,  tid, nth);
    cpy_lds(w + OFF_WK,  w_k,  16,  tid, nth);
    cpy_lds(w + OFF_WV,  w_v,  16,  tid, nth);
    cpy_lds(w + OFF_WFC, w_fc, 4,   tid, nth);
    cpy_lds(w + OFF_BFC, b_fc, 1,   tid, nth);
    __syncthreads();

    // ---- per-token MLP + q/k/v (threads 0..127, weights broadcast from LDS) ----
    if (tid < SEQ) {
        const float4* xp = (const float4*)(x + ((size_t)b * SEQ + tid) * 8);
        float4 a0 = xp[0], a1 = xp[1];
        float xin[8] = {a0.x, a0.y, a0.z, a0.w, a1.x, a1.y, a1.z, a1.w};

        float h1[16];
        #pragma unroll
        for (int o = 0; o < 16; ++o) {
            float acc = w[OFF_BFM + o];
            #pragma unroll
            for (int i = 0; i < 8; ++i) acc += w[OFF_WFM + o * 8 + i] * xin[i];
            h1[o] = tanhf(acc);
        }
        float h2[16];
        #pragma unroll
        for (int o = 0; o < 16; ++o) {
            float acc = w[OFF_BC1 + o];
            #pragma unroll
            for (int i = 0; i < 16; ++i) acc += w[OFF_WC1 + o * 16 + i] * h1[i];
            h2[o] = tanhf(acc);
        }
        float h3[12];
        #pragma unroll
        for (int o = 0; o < 12; ++o) {
            float acc = w[OFF_BP1 + o];
            #pragma unroll
            for (int i = 0; i < 16; ++i) acc += w[OFF_WP1 + o * 16 + i] * h2[i];
            h3[o] = tanhf(acc);
        }
        float h4[8];
        #pragma unroll
        for (int o = 0; o < 8; ++o) {
            float acc = w[OFF_BC2 + o];
            #pragma unroll
            for (int i = 0; i < 12; ++i) acc += w[OFF_WC2 + o * 12 + i] * h3[i];
            h4[o] = tanhf(acc);
        }
        float h5[4];
        #pragma unroll
        for (int o = 0; o < 4; ++o) {
            float acc = w[OFF_BP2 + o];
            #pragma unroll
            for (int i = 0; i < 8; ++i) acc += w[OFF_WP2 + o * 8 + i] * h4[i];
            h5[o] = tanhf(acc);
        }
        float h6[4];
        #pragma unroll
        for (int o = 0; o < 4; ++o) {
            float acc = w[OFF_BC3 + o];
            #pragma unroll
            for (int i = 0; i < 4; ++i) acc += w[OFF_WC3 + o * 4 + i] * h5[i];
            h6[o] = tanhf(acc);
        }
        #pragma unroll
        for (int o = 0; o < 4; ++o) {
            float aq = 0.f, ak = 0.f, av = 0.f;
            #pragma unroll
            for (int i = 0; i < 4; ++i) {
                aq += w[OFF_WQ + o * 4 + i] * h6[i];
                ak += w[OFF_WK + o * 4 + i] * h6[i];
                av += w[OFF_WV + o * 4 + i] * h6[i];
            }
            qs[tid][o] = aq; ks[tid][o] = ak; vs[tid][o] = av;
        }
    }
    __syncthreads();

    // ---- scores = (Q K^T) * 0.5 via v_wmma_f32_16x16x4_f32 ----
    // wave w owns query rows [16w, 16w+16); 8 WMMAs sweep all key tiles.
    {
        const int wave  = tid >> 5;          // 0..7
        const int lane  = tid & 31;
        const int m     = lane & 15;         // row (A) / col (B)
        const int khalf = (lane >> 4) * 2;   // K = khalf, khalf+1

        v2f A;
        A.x = qs[wave * 16 + m][khalf];
        A.y = qs[wave * 16 + m][khalf + 1];

        #pragma unroll
        for (int t = 0; t < 8; ++t) {
            v2f Bm;
            Bm.x = ks[t * 16 + m][khalf];
            Bm.y = ks[t * 16 + m][khalf + 1];
            v8f C = {};
            C = __builtin_amdgcn_wmma_f32_16x16x4_f32(
                    /*neg_a=*/false, A, /*neg_b=*/false, Bm,
                    /*c_mod=*/(short)0, C, /*reuse_a=*/false, /*reuse_b=*/false);
            #pragma unroll
            for (int r = 0; r < 8; ++r) {
                int row = wave * 16 + r + 8 * (lane >> 4);
                int col = t * 16 + m;
                sc[row][col] = C[r] * 0.5f;   // 1/sqrt(4)
            }
        }
    }
    __syncthreads();

    // ---- softmax fused with attn@V and final fc+sigmoid (threads 0..127) ----
    if (tid < SEQ) {
        float mx = -INFINITY;
        for (int k = 0; k < SEQ; ++k) mx = fmaxf(mx, sc[tid][k]);
        float sum = 0.f;
        float acc0 = 0.f, acc1 = 0.f, acc2 = 0.f, acc3 = 0.f;
        for (int k = 0; k < SEQ; ++k) {
            float p = __expf(sc[tid][k] - mx);
            sum += p;
            acc0 += p * vs[k][0];
            acc1 += p * vs[k][1];
            acc2 += p * vs[k][2];
            acc3 += p * vs[k][3];
        }
        float inv = 1.f / sum;
        float o = w[OFF_BFC]
                + w[OFF_WFC + 0] * acc0 * inv
                + w[OFF_WFC + 1] * acc1 * inv
                + w[OFF_WFC + 2] * acc2 * inv
                + w[OFF_WFC + 3] * acc3 * inv;
        out[(size_t)b * SEQ + tid] = 1.f / (1.f + __expf(-o));
    }
}

extern "C" void kernel_launch(void* const* d_in, const int* in_sizes, int n_in,
                              void* d_out, int out_size, void* d_ws, size_t ws_size,
                              hipStream_t stream) {
    const float* x    = (const float*)d_in[0];
    const float* w_fm = (const float*)d_in[1];
    const float* b_fm = (const float*)d_in[2];
    const float* w_c1 = (const float*)d_in[3];
    const float* b_c1 = (const float*)d_in[4];
    const float* w_p1 = (const float*)d_in[5];
    const float* b_p1 = (const float*)d_in[6];
    const float* w_c2 = (const float*)d_in[7];
    const float* b_c2 = (const float*)d_in[8];
    const float* w_p2 = (const float*)d_in[9];
    const float* b_p2 = (const float*)d_in[10];
    const float* w_c3 = (const float*)d_in[11];
    const float* b_c3 = (const float*)d_in[12];
    const float* w_q  = (const float*)d_in[13];
    const float* w_k  = (const float*)d_in[14];
    const float* w_v  = (const float*)d_in[15];
    const float* w_fc = (const float*)d_in[16];
    const float* b_fc = (const float*)d_in[17];

    const int B = in_sizes[0] / (SEQ * 8);   // 2048
    dim3 grid(B), block(256);
    qcnn_fused_kernel<<<grid, block, 0, stream>>>(
        x, w_fm, b_fm, w_c1, b_c1, w_p1, b_p1, w_c2, b_c2,
        w_p2, b_p2, w_c3, b_c3, w_q, w_k, w_v, w_fc, b_fc,
        (float*)d_out);
}